// RGCNGraphClassifier_18545668784637
// MI455X (gfx1250) — compile-verified
//
#include <hip/hip_runtime.h>
#include <hip/hip_bf16.h>

// ---------------------------------------------------------------------------
// RGCN graph classifier for MI455X (gfx1250, wave32).
//
// Math restructuring: per-relation mean aggregation commutes with the linear
// relation transform, so we scatter raw node features into S[R,N,H] (+counts),
// then do 9 dense fp32 WMMA GEMMs per layer:
//     y = relu( sum_r (S[r]/cnt[r]) @ W[r]  +  x @ W_root + b )
// Precision: full fp32 via V_WMMA_F32_16X16X4_F32 (16x16 f32 acc, K=4).
// Data path: Tensor Data Mover stages the active 128x128 weight matrix into
// LDS (one tensor_load_to_lds per matrix), B fed from LDS, A streamed from
// global with the mean-scale folded into the load.
// ---------------------------------------------------------------------------

typedef __attribute__((ext_vector_type(2))) float v2f;
typedef __attribute__((ext_vector_type(8))) float v8f;
typedef __attribute__((ext_vector_type(4))) unsigned int v4u;
typedef __attribute__((ext_vector_type(8))) int v8i;
typedef __attribute__((ext_vector_type(4))) int v4i;

#define HDIM 128
#define RREL 8
#define BGRAPH 64

__device__ __forceinline__ void atomAddF(float* p, float v) {
  // hardware global_atomic_add_f32 (no CAS loop)
  unsafeAtomicAdd(p, v);
}

// ------------------------------------------------------------------ TDM -----
// Issue one Tensor-DMA: load a 128x128 f32 row-major matrix from global into
// LDS at byte offset lds_off.  D# per CDNA5 ISA ch.8:
//   group0: count=1 | lds_addr | global_addr[56:0] | type=2
//   group1: data_size=4B, tensor_dim0=128, tensor_dim1=128,
//           tile_dim0=128, tile_dim1=128, tensor_dim0_stride=128
//   groups 2/3: zero (2-D tensor)
__device__ __forceinline__ void tdm_load_w(unsigned lds_off, const float* gptr) {
  unsigned long long ga = (unsigned long long)(uintptr_t)gptr;
  v4u g0;
  g0[0] = 1u;                                        // count=1, user mode
  g0[1] = lds_off;                                   // lds_addr (bytes)
  g0[2] = (unsigned)(ga & 0xFFFFFFFFu);              // global_addr[31:0]
  g0[3] = (unsigned)((ga >> 32) & 0x01FFFFFFu)       // global_addr[56:32]
          | (2u << 30);                              // type=2 ("image")
  v8i g1;
  g1[0] = 0x00020000;          // workgroup_mask=0, data_size=2 (4 bytes)
  g1[1] = (HDIM & 0xFFFF) << 16;  // tensor_dim0[15:0] @ bits 63:48
  g1[2] = (HDIM & 0xFFFF) << 16;  // tensor_dim1[15:0] @ bits 95:80
  g1[3] = (HDIM & 0xFFFF) << 16;  // tile_dim0 @ bits 127:112
  g1[4] = HDIM;                   // tile_dim1 @ bits 143:128, tile_dim2=0
  g1[5] = HDIM;                   // tensor_dim0_stride[31:0]
  g1[6] = (int)((HDIM * HDIM) & 0xFFFF) << 16;  // tensor_dim1_stride[15:0]
  g1[7] = 0;                      // tensor_dim1_stride[47:16]
#if defined(__clang_major__) && (__clang_major__ >= 23)
  __builtin_amdgcn_tensor_load_to_lds(g0, g1, (v4i)0, (v4i)0, (v8i)0, 0);
#else
  __builtin_amdgcn_tensor_load_to_lds(g0, g1, (v4i)0, (v4i)0, 0);
#endif
}

// ---------------------------------------------------------------- zero ------
__global__ void zero_k(float4* __restrict__ p, long n4) {
  long i = (long)blockIdx.x * blockDim.x + threadIdx.x;
  long stride = (long)gridDim.x * blockDim.x;
  float4 z = make_float4(0.f, 0.f, 0.f, 0.f);
  for (; i < n4; i += stride) p[i] = z;
}

// ---------------------------------------------------------------- embed -----
// x[n,:] = emb[node_type[n],:]   (one wave per node, float4 per lane)
__global__ void embed_k(const int* __restrict__ nt, const float* __restrict__ emb,
                        float* __restrict__ x) {
  int tid = blockIdx.x * blockDim.x + threadIdx.x;
  int n = tid >> 5, lane = tid & 31;
  int t = nt[n];
  ((float4*)(x + (size_t)n * HDIM))[lane] =
      ((const float4*)(emb + (size_t)t * HDIM))[lane];
}

// --------------------------------------------------------------- scatter ----
// S[r,dst,:] += x[src,:] ; cnt[r,dst] += 1      (one wave per edge)
__global__ void scatter_k(const float* __restrict__ x, const int* __restrict__ src,
                          const int* __restrict__ dst, const int* __restrict__ et,
                          float* __restrict__ S, float* __restrict__ cnt, int N) {
  int tid = blockIdx.x * blockDim.x + threadIdx.x;
  int e = tid >> 5, lane = tid & 31;
  int s = src[e], d = dst[e], r = et[e];
  float4 v = ((const float4*)(x + (size_t)s * HDIM))[lane];
  float* p = S + ((size_t)r * N + d) * HDIM + lane * 4;
  atomAddF(p + 0, v.x);
  atomAddF(p + 1, v.y);
  atomAddF(p + 2, v.z);
  atomAddF(p + 3, v.w);
  if (lane == 0) atomAddF(cnt + (size_t)r * N + d, 1.0f);
}

// ----------------------------------------------------------------- gemm -----
// One wave computes a 16-row x 128-col output tile:
//   y[tile] = relu( sum_{r} (S[r,tile]*sc) @ Wrel[r] + x[tile] @ Wroot + b )
// Weight matrix staged in LDS by TDM (wave 0 issues, TENSORcnt wait, barrier).
// A (16x4 f32) per ISA: lanes0-15 M=0..15 {K0,K1}, lanes16-31 {K2,K3}; read
// straight from global with 1/cnt folded in (scale depends only on lane&15).
// B (4x16 f32) mirrored: lanes0-15 rows k,k+1; lanes16-31 rows k+2,k+3.
// D (16x16 f32): VGPR i -> M=i (lanes0-15) / M=i+8 (lanes16-31), N = lane&15.
__global__ __launch_bounds__(128)
void rgcn_gemm_k(const float* __restrict__ x, const float* __restrict__ S,
                 const float* __restrict__ cnt, const float* __restrict__ Wrel,
                 const float* __restrict__ Wroot, const float* __restrict__ bias,
                 float* __restrict__ y, int N) {
  __shared__ float WL[HDIM * HDIM];  // 64 KB, filled by the Tensor Data Mover
  const int lane  = threadIdx.x & 31;
  const int wave  = threadIdx.x >> 5;
  const int tile  = blockIdx.x * 4 + wave;
  const int row0  = tile * 16;
  const int khalf = (lane >> 4) * 2;  // 0 or 2
  const int l15   = lane & 15;
  const unsigned wl_off = (unsigned)(uintptr_t)(void*)&WL[0];  // LDS byte addr

  v8f acc[8];
#pragma unroll
  for (int i = 0; i < 8; ++i) acc[i] = (v8f)0.f;

  for (int m = 0; m < RREL + 1; ++m) {
    // ---- stage W[m] into LDS via TDM (wave 0 only; per-wave issue) ----
    const float* Wm =
        (m < RREL) ? (Wrel + (size_t)m * HDIM * HDIM) : Wroot;
    if (threadIdx.x < 32) {
      tdm_load_w(wl_off, Wm);
      __builtin_amdgcn_s_wait_tensorcnt(0);
    }
    // ---- per-lane A row pointer + mean scale ----
    const float* arow;
    float sc;
    if (m < RREL) {
      float c = cnt[(size_t)m * N + row0 + l15];
      sc = (c > 0.f) ? (1.0f / c) : 0.0f;  // mean with empty-segment mask
      arow = S + ((size_t)m * N + row0 + l15) * HDIM;
    } else {
      sc = 1.0f;
      arow = x + (size_t)(row0 + l15) * HDIM;
    }
    __syncthreads();  // WL visible to all 4 waves

    for (int k0 = 0; k0 < HDIM; k0 += 4) {
      const int ka = k0 + khalf;
      v2f a;
      a.x = arow[ka] * sc;
      a.y = arow[ka + 1] * sc;
#pragma unroll
      for (int n0 = 0; n0 < 8; ++n0) {
        const int col = n0 * 16 + l15;
        v2f b;
        b.x = WL[ka * HDIM + col];
        b.y = WL[(ka + 1) * HDIM + col];
        acc[n0] = __builtin_amdgcn_wmma_f32_16x16x4_f32(
            false, a, false, b, (short)0, acc[n0], false, false);
      }
    }
    __syncthreads();  // all waves done reading WL before next TDM overwrite
  }

  // epilogue: bias + relu + store
#pragma unroll
  for (int i = 0; i < 8; ++i) {
    int rrow = row0 + i + (lane >> 4) * 8;
#pragma unroll
    for (int n0 = 0; n0 < 8; ++n0) {
      int col = n0 * 16 + l15;
      float v = acc[n0][i] + bias[col];
      y[(size_t)rrow * HDIM + col] = fmaxf(v, 0.f);
    }
  }
}

// ----------------------------------------------------------------- pool -----
// gsum[batch[n],:] += x[n,:] ; gcnt[batch[n]] += 1   (one wave per node)
__global__ void pool_k(const float* __restrict__ x, const int* __restrict__ batch,
                       float* __restrict__ gsum, float* __restrict__ gcnt) {
  int tid = blockIdx.x * blockDim.x + threadIdx.x;
  int n = tid >> 5, lane = tid & 31;
  int b = batch[n];
  float4 v = ((const float4*)(x + (size_t)n * HDIM))[lane];
  float* p = gsum + (size_t)b * HDIM + lane * 4;
  atomAddF(p + 0, v.x);
  atomAddF(p + 1, v.y);
  atomAddF(p + 2, v.z);
  atomAddF(p + 3, v.w);
  if (lane == 0) atomAddF(gcnt + b, 1.0f);
}

// -------------------------------------------------------------- classify ----
// g = gsum/max(gcnt,1); h = relu(g@W1+b1); out = h@W2+b2   (single block)
__global__ __launch_bounds__(128)
void classify_k(const float* __restrict__ gsum, const float* __restrict__ gcnt,
                const float* __restrict__ W1, const float* __restrict__ b1,
                const float* __restrict__ W2, const float* __restrict__ b2,
                float* __restrict__ out) {
  __shared__ float g[BGRAPH * HDIM];
  __shared__ float h[BGRAPH * HDIM];
  int t = threadIdx.x;
  for (int i = t; i < BGRAPH * HDIM; i += 128) {
    int b = i >> 7;
    g[i] = gsum[i] / fmaxf(gcnt[b], 1.0f);
  }
  __syncthreads();
  for (int i = t; i < BGRAPH * HDIM; i += 128) {
    int b = i >> 7, o = i & 127;
    float s = b1[o];
    for (int k = 0; k < HDIM; ++k) s += g[b * HDIM + k] * W1[k * HDIM + o];
    h[i] = fmaxf(s, 0.f);
  }
  __syncthreads();
  // B*C = 128 outputs, exactly one per thread
  {
    int b = t >> 1, c = t & 1;
    float s = b2[c];
    for (int k = 0; k < HDIM; ++k) s += h[b * HDIM + k] * W2[k * 2 + c];
    out[t] = s;
  }
}

// ---------------------------------------------------------------- launch ----
extern "C" void kernel_launch(void* const* d_in, const int* in_sizes, int n_in,
                              void* d_out, int out_size, void* d_ws, size_t ws_size,
                              hipStream_t stream) {
  const int* node_type = (const int*)d_in[0];
  const int* edge_src  = (const int*)d_in[1];
  const int* edge_dst  = (const int*)d_in[2];
  const int* edge_type = (const int*)d_in[3];
  const int* batch     = (const int*)d_in[4];
  const float* emb        = (const float*)d_in[5];
  const float* conv0_W    = (const float*)d_in[6];
  const float* conv0_root = (const float*)d_in[7];
  const float* conv0_b    = (const float*)d_in[8];
  const float* conv1_W    = (const float*)d_in[9];
  const float* conv1_root = (const float*)d_in[10];
  const float* conv1_b    = (const float*)d_in[11];
  const float* cls_W1 = (const float*)d_in[12];
  const float* cls_b1 = (const float*)d_in[13];
  const float* cls_W2 = (const float*)d_in[14];
  const float* cls_b2 = (const float*)d_in[15];
  float* out = (float*)d_out;

  const int N = in_sizes[0];
  const int E = in_sizes[1];

  // workspace layout (floats):
  //   S [R*N*H] | cnt [R*N] | x [N*H] | x2 [N*H] | gsum [B*H] | gcnt [B]
  float* ws = (float*)d_ws;
  const size_t nS = (size_t)RREL * N * HDIM;
  const size_t nC = (size_t)RREL * N;
  const size_t nX = (size_t)N * HDIM;
  float* S    = ws;
  float* cnt  = S + nS;
  float* x    = cnt + nC;
  float* x2   = x + nX;
  float* gsum = x2 + nX;
  float* gcnt = gsum + (size_t)BGRAPH * HDIM;

  const int ZB = 2048, ZT = 256;
  const size_t gsz = (size_t)BGRAPH * HDIM + BGRAPH;  // gsum+gcnt, mult of 4

  // ---- layer 0 ----
  zero_k<<<ZB, ZT, 0, stream>>>((float4*)S, (long)((nS + nC) / 4));
  zero_k<<<64, ZT, 0, stream>>>((float4*)gsum, (long)(gsz / 4));
  embed_k<<<(N * 32) / 256, 256, 0, stream>>>(node_type, emb, x);
  scatter_k<<<(E * 32) / 256, 256, 0, stream>>>(x, edge_src, edge_dst, edge_type,
                                                S, cnt, N);
  rgcn_gemm_k<<<N / 64, 128, 0, stream>>>(x, S, cnt, conv0_W, conv0_root,
                                          conv0_b, x2, N);
  // ---- layer 1 ----
  zero_k<<<ZB, ZT, 0, stream>>>((float4*)S, (long)((nS + nC) / 4));
  scatter_k<<<(E * 32) / 256, 256, 0, stream>>>(x2, edge_src, edge_dst, edge_type,
                                                S, cnt, N);
  rgcn_gemm_k<<<N / 64, 128, 0, stream>>>(x2, S, cnt, conv1_W, conv1_root,
                                          conv1_b, x, N);
  // ---- readout ----
  pool_k<<<(N * 32) / 256, 256, 0, stream>>>(x, batch, gsum, gcnt);
  classify_k<<<1, 128, 0, stream>>>(gsum, gcnt, cls_W1, cls_b1, cls_W2, cls_b2,
                                    out);
}